// SpatialEmbedding_48670569398942
// MI455X (gfx1250) — compile-verified
//
#include <hip/hip_runtime.h>
#include <hip/hip_bf16.h>
#include <math.h>

typedef float v2f __attribute__((ext_vector_type(2)));
typedef float v8f __attribute__((ext_vector_type(8)));

#define LDK 36            // padded LDS K-stride: 16B-aligned rows, conflict-free frag loads
#define BN_EPS 1e-5f

// ---------------------------------------------------------------------------
// conv1: h1[m][o] = xyz[m] . w[o]   (K=3, O=64)
// ---------------------------------------------------------------------------
__global__ __launch_bounds__(256)
void k_conv1(const float* __restrict__ xyz, const float* __restrict__ w,
             float* __restrict__ h1, int total) {
  int g = blockIdx.x * 256 + threadIdx.x;
  if (g >= total) return;
  int m = g >> 6, o = g & 63;
  const float* p  = xyz + m * 3;
  const float* ww = w + o * 3;
  h1[g] = p[0] * ww[0] + p[1] * ww[1] + p[2] * ww[2];
}

// ---------------------------------------------------------------------------
// per-channel sum / sumsq over rows (column reduction with atomics)
// ---------------------------------------------------------------------------
__global__ __launch_bounds__(256)
void k_colstats(const float* __restrict__ H, float* __restrict__ sum,
                float* __restrict__ ss, int M, int O, int mPerBlock) {
  int o = blockIdx.x * 256 + threadIdx.x;
  if (o >= O) return;
  int m0 = blockIdx.y * mPerBlock;
  int m1 = m0 + mPerBlock; if (m1 > M) m1 = M;
  float s = 0.f, s2 = 0.f;
  for (int m = m0; m < m1; ++m) {
    float v = H[(size_t)m * O + o];
    s += v; s2 += v * v;
  }
  atomicAdd(&sum[o], s);
  atomicAdd(&ss[o], s2);
}

// fold BN into per-channel affine: y = a*x + c
__global__ void k_finalize(const float* __restrict__ sum, const float* __restrict__ ss,
                           const float* __restrict__ gamma, const float* __restrict__ beta,
                           float* __restrict__ affA, float* __restrict__ affC,
                           int O, float invCnt) {
  int o = blockIdx.x * blockDim.x + threadIdx.x;
  if (o >= O) return;
  float mean = sum[o] * invCnt;
  float var  = ss[o] * invCnt - mean * mean;
  float inv  = rsqrtf(var + BN_EPS);
  float a    = gamma[o] * inv;
  affA[o] = a;
  affC[o] = beta[o] - a * mean;
}

// elementwise relu(a*x+c) (materialize feat256)
__global__ __launch_bounds__(256)
void k_bnrelu(const float* __restrict__ H, const float* __restrict__ a,
              const float* __restrict__ c, float* __restrict__ out,
              int C, int total) {
  int g = blockIdx.x * 256 + threadIdx.x;
  if (g >= total) return;
  int ch = g % C;
  out[g] = fmaxf(0.f, a[ch] * H[g] + c[ch]);
}

// ---------------------------------------------------------------------------
// top-8 nearest neighbors per point (shared across both spa_group calls)
// ---------------------------------------------------------------------------
__global__ __launch_bounds__(256)
void k_topk(const float* __restrict__ xyz, int* __restrict__ idx, int Nn) {
  __shared__ float xs[2048], ys[2048], zs[2048];
  int blocksPerB = Nn / 256;
  int b = blockIdx.x / blocksPerB;
  int blkInB = blockIdx.x % blocksPerB;
  const float* xb = xyz + (size_t)b * Nn * 3;
  for (int i = threadIdx.x; i < Nn; i += 256) {
    xs[i] = xb[i * 3 + 0];
    ys[i] = xb[i * 3 + 1];
    zs[i] = xb[i * 3 + 2];
  }
  __syncthreads();
  int n = blkInB * 256 + threadIdx.x;
  float qx = xs[n], qy = ys[n], qz = zs[n];
  float bd[8]; int bi[8];
#pragma unroll
  for (int j = 0; j < 8; ++j) { bd[j] = 3.4e38f; bi[j] = 0; }
  for (int m = 0; m < Nn; ++m) {
    float dx = qx - xs[m], dy = qy - ys[m], dz = qz - zs[m];
    float d = dx * dx + dy * dy + dz * dz;
    if (d < bd[7]) {
      bd[7] = d; bi[7] = m;
#pragma unroll
      for (int j = 7; j > 0; --j) {
        if (bd[j] < bd[j - 1]) {
          float td = bd[j]; bd[j] = bd[j - 1]; bd[j - 1] = td;
          int ti = bi[j];  bi[j] = bi[j - 1];  bi[j - 1] = ti;
        }
      }
    }
  }
  int* op = idx + ((size_t)b * Nn + n) * 8;
#pragma unroll
  for (int j = 0; j < 8; ++j) op[j] = bi[j];
}

// ---------------------------------------------------------------------------
// WMMA fp32 GEMM: Hout[M][O] = A[M][K] * W[O][K]^T
// MODE 0: A[m][k] = relu(affA[k]*Asrc[m][k] + affC[k])
// MODE 1: A[m][k] = k<C ? feat[src][k]-feat[ctr][k] : feat[ctr][k-C]
// 256 thr / 8 waves; block tile 128x128, BK=32; wave tile 32x64 (8 x v8f acc).
// Double-buffered LDS; W tile staged via GLOBAL_LOAD_ASYNC_TO_LDS_B128
// (ASYNCcnt), A tile software-pipelined through registers (transform fused).
// ---------------------------------------------------------------------------
template <int MODE>
__global__ __launch_bounds__(256)
void k_gemm_wmma(const float* __restrict__ Asrc, const float* __restrict__ W,
                 float* __restrict__ Hout,
                 const float* __restrict__ affA, const float* __restrict__ affC,
                 const int* __restrict__ idx,
                 int M, int K, int O, int Cfeat, int Nn) {
  __shared__ float As[2][128 * LDK];
  __shared__ float Ws[2][128 * LDK];

  const int t = threadIdx.x;
  const int lane = t & 31;
  const int wv = t >> 5;
  const int m0 = blockIdx.y * 128;
  const int o0 = blockIdx.x * 128;
  const int rb = (wv & 3) * 32;   // wave row strip (32 rows)
  const int cb = (wv >> 2) * 64;  // wave col strip (64 cols)

  v8f acc[2][4];
#pragma unroll
  for (int r = 0; r < 2; ++r)
#pragma unroll
    for (int c2 = 0; c2 < 4; ++c2)
      acc[r][c2] = v8f{0.f, 0.f, 0.f, 0.f, 0.f, 0.f, 0.f, 0.f};

  // A staging coords: row li, k-half lh (16 floats per thread)
  const int li = t >> 1;
  const int lh = (t & 1) * 16;

  const float* frow_src = nullptr;
  const float* frow_ctr = nullptr;
  if (MODE == 1) {
    int m = m0 + li;
    int m8 = m >> 3;
    int gb = m8 / Nn;
    int gn = m8 % Nn;
    int src = idx[m];
    frow_src = Asrc + ((size_t)(gb * Nn + src)) * Cfeat;
    frow_ctr = Asrc + ((size_t)(gb * Nn + gn)) * Cfeat;
  }

  float4 rA[4];

  // load + transform A chunk for tile at k0 into registers
  auto loadA = [&](int k0) {
    if (MODE == 0) {
      const float* hp = Asrc + (size_t)(m0 + li) * K + k0 + lh;
#pragma unroll
      for (int q = 0; q < 4; ++q) {
        float4 hv = *(const float4*)(hp + q * 4);
        float4 av = *(const float4*)(affA + k0 + lh + q * 4);
        float4 cv = *(const float4*)(affC + k0 + lh + q * 4);
        rA[q].x = fmaxf(0.f, av.x * hv.x + cv.x);
        rA[q].y = fmaxf(0.f, av.y * hv.y + cv.y);
        rA[q].z = fmaxf(0.f, av.z * hv.z + cv.z);
        rA[q].w = fmaxf(0.f, av.w * hv.w + cv.w);
      }
    } else {
      if (k0 < Cfeat) {
        const float* ps = frow_src + k0 + lh;
        const float* pc = frow_ctr + k0 + lh;
#pragma unroll
        for (int q = 0; q < 4; ++q) {
          float4 sv = *(const float4*)(ps + q * 4);
          float4 cv = *(const float4*)(pc + q * 4);
          rA[q].x = sv.x - cv.x;
          rA[q].y = sv.y - cv.y;
          rA[q].z = sv.z - cv.z;
          rA[q].w = sv.w - cv.w;
        }
      } else {
        const float* pc = frow_ctr + (k0 - Cfeat) + lh;
#pragma unroll
        for (int q = 0; q < 4; ++q) rA[q] = *(const float4*)(pc + q * 4);
      }
    }
  };

  auto storeA = [&](int buf) {
    float* dst = &As[buf][li * LDK + lh];
#pragma unroll
    for (int q = 0; q < 4; ++q) *(float4*)(dst + q * 4) = rA[q];
  };

  // async copy of W tile (pure memcpy): 1024 16B chunks, 4 per thread
  auto asyncW = [&](int k0, int buf) {
#pragma unroll
    for (int i = 0; i < 4; ++i) {
      int cid = t + 256 * i;
      int row = cid >> 3;
      int off = (cid & 7) * 4;
      const float* gp = W + (size_t)(o0 + row) * K + k0 + off;
      unsigned ldsa = (unsigned)(size_t)&Ws[buf][row * LDK + off];
      asm volatile("global_load_async_to_lds_b128 %0, %1, off"
                   :: "v"(ldsa), "v"((unsigned long long)(size_t)gp)
                   : "memory");
    }
  };

  const int NT = K / 32;

  // prologue: stage tile 0
  loadA(0);
  asyncW(0, 0);
  storeA(0);
  asm volatile("s_wait_asynccnt 0" ::: "memory");
  __syncthreads();

  const int ml = lane & 15;
  const int kh = 2 * (lane >> 4);

  for (int kt = 0; kt < NT; ++kt) {
    const int cur = kt & 1;
    const int nxt = cur ^ 1;
    if (kt + 1 < NT) {
      asyncW((kt + 1) * 32, nxt);   // writes buffer last read 2 tiles ago (barrier-separated)
      loadA((kt + 1) * 32);
    }

    // ---- compute on buffer `cur`: 8 k-steps of 4 ----
#pragma unroll
    for (int kk = 0; kk < 32; kk += 4) {
      const int kl = kk + kh;
      v2f afr[2], wfr[4];
#pragma unroll
      for (int r = 0; r < 2; ++r)
        afr[r] = *(const v2f*)&As[cur][(rb + r * 16 + ml) * LDK + kl];
#pragma unroll
      for (int c2 = 0; c2 < 4; ++c2)
        wfr[c2] = *(const v2f*)&Ws[cur][(cb + c2 * 16 + ml) * LDK + kl];
#pragma unroll
      for (int r = 0; r < 2; ++r)
#pragma unroll
        for (int c2 = 0; c2 < 4; ++c2)
          acc[r][c2] = __builtin_amdgcn_wmma_f32_16x16x4_f32(
              false, afr[r], false, wfr[c2], (short)0, acc[r][c2], false, false);
    }

    if (kt + 1 < NT) {
      storeA(nxt);
      asm volatile("s_wait_asynccnt 0" ::: "memory");
      __syncthreads();
    }
  }

  // ---- epilogue: raw output (BN stats reduced in a later pass) ----
  const int hi = (lane >> 4) * 8;
#pragma unroll
  for (int r = 0; r < 2; ++r)
#pragma unroll
    for (int c2 = 0; c2 < 4; ++c2) {
      int og = o0 + cb + c2 * 16 + ml;
#pragma unroll
      for (int j = 0; j < 8; ++j) {
        int m = m0 + rb + r * 16 + hi + j;
        Hout[(size_t)m * O + og] = acc[r][c2][j];
      }
    }
}

// ---------------------------------------------------------------------------
// relu(a*x+c) then max over k=8 rows; optional transposed write (B,O,N)
// ---------------------------------------------------------------------------
__global__ __launch_bounds__(256)
void k_maxk(const float* __restrict__ H, const float* __restrict__ a,
            const float* __restrict__ c, float* __restrict__ out,
            int O, int Nn, int transpose, int total) {
  int g = blockIdx.x * 256 + threadIdx.x;
  if (g >= total) return;
  int o = g % O;
  int bn = g / O;
  const float* hp = H + (size_t)bn * 8 * O + o;
  float av = a[o], cv = c[o];
  float v = 0.f;  // max_k relu(x) == max(0, max_k x)
#pragma unroll
  for (int kk = 0; kk < 8; ++kk) v = fmaxf(v, av * hp[(size_t)kk * O] + cv);
  if (transpose) {
    int b = bn / Nn, n = bn % Nn;
    out[((size_t)b * O + o) * Nn + n] = v;
  } else {
    out[(size_t)bn * O + o] = v;
  }
}

// ---------------------------------------------------------------------------
extern "C" void kernel_launch(void* const* d_in, const int* in_sizes, int n_in,
                              void* d_out, int out_size, void* d_ws, size_t ws_size,
                              hipStream_t stream) {
  const int B = 4, Nn = 2048, BN = B * Nn;       // 8192
  const int Mg = BN * 8;                          // 65536 grouped rows

  const float* xyz  = (const float*)d_in[0];
  const float* w_c1 = (const float*)d_in[1];
  const float* g_c1 = (const float*)d_in[2];
  const float* b_c1 = (const float*)d_in[3];
  const float* w_c2 = (const float*)d_in[4];
  const float* g_c2 = (const float*)d_in[5];
  const float* b_c2 = (const float*)d_in[6];
  const float* w_l0a = (const float*)d_in[7];
  const float* g_l0a = (const float*)d_in[8];
  const float* b_l0a = (const float*)d_in[9];
  const float* w_l0b = (const float*)d_in[10];
  const float* g_l0b = (const float*)d_in[11];
  const float* b_l0b = (const float*)d_in[12];
  const float* w_l1a = (const float*)d_in[13];
  const float* g_l1a = (const float*)d_in[14];
  const float* b_l1a = (const float*)d_in[15];
  const float* w_l1b = (const float*)d_in[16];
  const float* g_l1b = (const float*)d_in[17];
  const float* b_l1b = (const float*)d_in[18];
  // d_in[19] = k (==8, hardcoded)

  char* ws = (char*)d_ws;
  float* h1      = (float*)(ws);                               // 8192*64
  float* feat256 = (float*)(ws + (2ull  << 20));               // 8192*256
  float* feat512 = (float*)(ws + (11ull << 20));               // 8192*512
  int*   idxb    = (int*)  (ws + (28ull << 20));               // 8192*8
  float* sum     = (float*)(ws + (29ull << 20));               // 1024
  float* ss      = sum + 1024;
  float* affA    = ss + 1024;
  float* affC    = affA + 1024;
  float* hA      = (float*)(ws + (32ull  << 20));              // 65536*1024
  float* hB      = (float*)(ws + (320ull << 20));              // 65536*1024

  const size_t statBytes = 2 * 1024 * sizeof(float);

  // -------- conv1 (3 -> 64) --------
  k_conv1<<<(BN * 64) / 256, 256, 0, stream>>>(xyz, w_c1, h1, BN * 64);
  hipMemsetAsync(sum, 0, statBytes, stream);
  k_colstats<<<dim3(1, BN / 256), 256, 0, stream>>>(h1, sum, ss, BN, 64, 256);
  k_finalize<<<1, 64, 0, stream>>>(sum, ss, g_c1, b_c1, affA, affC, 64, 1.f / BN);

  // -------- conv2 (64 -> 256) via WMMA, A = relu(affine(h1)) --------
  k_gemm_wmma<0><<<dim3(256 / 128, BN / 128), 256, 0, stream>>>(
      h1, w_c2, hA, affA, affC, nullptr, BN, 64, 256, 0, Nn);
  hipMemsetAsync(sum, 0, statBytes, stream);
  k_colstats<<<dim3(1, BN / 256), 256, 0, stream>>>(hA, sum, ss, BN, 256, 256);
  k_finalize<<<1, 256, 0, stream>>>(sum, ss, g_c2, b_c2, affA, affC, 256, 1.f / BN);
  k_bnrelu<<<(BN * 256) / 256, 256, 0, stream>>>(hA, affA, affC, feat256, 256, BN * 256);

  // -------- kNN top-8 (shared by both grouping stages) --------
  k_topk<<<B * (Nn / 256), 256, 0, stream>>>(xyz, idxb, Nn);

  // -------- stage 0: gather(feat256) -> GEMM(512x512) x2 -> max_k --------
  k_gemm_wmma<1><<<dim3(512 / 128, Mg / 128), 256, 0, stream>>>(
      feat256, w_l0a, hA, nullptr, nullptr, idxb, Mg, 512, 512, 256, Nn);
  hipMemsetAsync(sum, 0, statBytes, stream);
  k_colstats<<<dim3(2, Mg / 256), 256, 0, stream>>>(hA, sum, ss, Mg, 512, 256);
  k_finalize<<<2, 256, 0, stream>>>(sum, ss, g_l0a, b_l0a, affA, affC, 512, 1.f / Mg);

  k_gemm_wmma<0><<<dim3(512 / 128, Mg / 128), 256, 0, stream>>>(
      hA, w_l0b, hB, affA, affC, nullptr, Mg, 512, 512, 0, Nn);
  hipMemsetAsync(sum, 0, statBytes, stream);
  k_colstats<<<dim3(2, Mg / 256), 256, 0, stream>>>(hB, sum, ss, Mg, 512, 256);
  k_finalize<<<2, 256, 0, stream>>>(sum, ss, g_l0b, b_l0b, affA, affC, 512, 1.f / Mg);
  k_maxk<<<(BN * 512) / 256, 256, 0, stream>>>(hB, affA, affC, feat512, 512, Nn, 0, BN * 512);

  // -------- stage 1: gather(feat512) -> GEMM(1024x1024) x2 -> max_k -> out^T --------
  k_gemm_wmma<1><<<dim3(1024 / 128, Mg / 128), 256, 0, stream>>>(
      feat512, w_l1a, hA, nullptr, nullptr, idxb, Mg, 1024, 1024, 512, Nn);
  hipMemsetAsync(sum, 0, statBytes, stream);
  k_colstats<<<dim3(4, Mg / 256), 256, 0, stream>>>(hA, sum, ss, Mg, 1024, 256);
  k_finalize<<<4, 256, 0, stream>>>(sum, ss, g_l1a, b_l1a, affA, affC, 1024, 1.f / Mg);

  k_gemm_wmma<0><<<dim3(1024 / 128, Mg / 128), 256, 0, stream>>>(
      hA, w_l1b, hB, affA, affC, nullptr, Mg, 1024, 1024, 0, Nn);
  hipMemsetAsync(sum, 0, statBytes, stream);
  k_colstats<<<dim3(4, Mg / 256), 256, 0, stream>>>(hB, sum, ss, Mg, 1024, 256);
  k_finalize<<<4, 256, 0, stream>>>(sum, ss, g_l1b, b_l1b, affA, affC, 1024, 1.f / Mg);
  k_maxk<<<(BN * 1024) / 256, 256, 0, stream>>>(hB, affA, affC, (float*)d_out, 1024, Nn, 1, BN * 1024);
}